// GNNLayer_46273977647662
// MI455X (gfx1250) — compile-verified
//
#include <hip/hip_runtime.h>
#include <hip/hip_bf16.h>

#define N_NODES 50000
#define N_EDGES 800000
#define D_DIM 64
#define M_STRIPS (N_NODES / 16)   // 3125 (exact)

typedef float v2f __attribute__((ext_vector_type(2)));
typedef float v8f __attribute__((ext_vector_type(8)));

// ---------------------------------------------------------------------------
// Kernel 1: zero the aggregation buffer (in workspace)
// ---------------------------------------------------------------------------
__global__ void gnn_zero_kernel(float4* __restrict__ agg4, int n4) {
    int i = blockIdx.x * 256 + threadIdx.x;
    if (i < n4) agg4[i] = make_float4(0.f, 0.f, 0.f, 0.f);
}

// ---------------------------------------------------------------------------
// Kernel 2: scatter-add  agg[dst[e]] += x[src[e]]
// 16 threads per edge, each handles 4 consecutive dims via a float4 gather
// + 4 global_atomic_add_f32 (resolved in L2; 12.8MB agg << 192MB L2).
// Indices are int64 with values < 50000 -> read only the low 32-bit word.
// ---------------------------------------------------------------------------
__global__ void gnn_scatter_kernel(const float* __restrict__ x,
                                   const unsigned int* __restrict__ src_lo,
                                   const unsigned int* __restrict__ dst_lo,
                                   float* __restrict__ agg,
                                   int total) {
    int i = blockIdx.x * 256 + threadIdx.x;
    if (i >= total) return;
    int e = i >> 4;              // edge id
    int d = (i & 15) << 2;       // starting dim (0,4,...,60)
    unsigned int s = src_lo[2 * e];   // low word of int64 (little-endian)
    unsigned int t = dst_lo[2 * e];
    const float4 v = *(const float4*)(x + (size_t)s * D_DIM + d);
    float* a = agg + (size_t)t * D_DIM + d;
    atomicAdd(a + 0, v.x);
    atomicAdd(a + 1, v.y);
    atomicAdd(a + 2, v.z);
    atomicAdd(a + 3, v.w);
}

// ---------------------------------------------------------------------------
// Kernel 3: out = relu( agg @ W_rel^T + b_rel + x @ W_root^T )
// One wave computes a full 16x64 output strip: A fragments for agg and x
// (16 k-steps each, v2f per lane) are loaded once into registers, then
// reused across the 4 N-tiles. 4*(16+16) = 128 V_WMMA_F32_16X16X4_F32 per
// wave, accumulating both GEMMs into one accumulator per N-tile.
//
// Fragment layouts (wave32, CDNA5 ISA):
//   A 16x4 f32: lane L holds A[L%16][(L/16)*2 + i], i=0,1  -> contiguous float2
//   B 4x16 f32: lane L holds B[(L/16)*2 + i][L%16]; B = W^T and W is stored
//               [out,in], so B[k][n] = W[n*64+k] -> contiguous float2
//   C/D 16x16:  vgpr r, lanes 0-15 -> M=r, lanes 16-31 -> M=r+8, N=L%16
// ---------------------------------------------------------------------------
__global__ void gnn_wmma_gemm_kernel(const float* __restrict__ agg,
                                     const float* __restrict__ x,
                                     const float* __restrict__ Wrel,
                                     const float* __restrict__ brel,
                                     const float* __restrict__ Wroot,
                                     float* __restrict__ out) {
    const int lane = threadIdx.x & 31;
    const int wave = threadIdx.x >> 5;
    const int strip = blockIdx.x * 8 + wave;
    if (strip >= M_STRIPS) return;       // wave-uniform: EXEC stays all-1s

    const int m0 = strip * 16;
    const int l16 = lane & 15;
    const int khalf = (lane >> 4) * 2;   // 0 or 2

    // Load all A fragments for this strip once (reused by all 4 N-tiles).
    const float* arow = agg + (size_t)(m0 + l16) * D_DIM + khalf;
    const float* xrow = x   + (size_t)(m0 + l16) * D_DIM + khalf;
    v2f a_agg[16];
    v2f a_x[16];
#pragma unroll
    for (int kk = 0; kk < 16; ++kk) {
        a_agg[kk] = *(const v2f*)(arow + kk * 4);
        a_x[kk]   = *(const v2f*)(xrow + kk * 4);
    }

    const int mbase = m0 + (lane >> 4) * 8;

#pragma unroll
    for (int tn = 0; tn < 4; ++tn) {
        const int n0 = tn * 16;
        const float* wrel  = Wrel  + (size_t)(n0 + l16) * D_DIM + khalf;
        const float* wroot = Wroot + (size_t)(n0 + l16) * D_DIM + khalf;

        v8f acc = {};
        // agg @ W_rel^T
#pragma unroll
        for (int kk = 0; kk < 16; ++kk) {
            v2f b = *(const v2f*)(wrel + kk * 4);
            acc = __builtin_amdgcn_wmma_f32_16x16x4_f32(
                /*neg_a=*/false, a_agg[kk], /*neg_b=*/false, b,
                /*c_mod=*/(short)0, acc, /*reuse_a=*/false, /*reuse_b=*/false);
        }
        // + x @ W_root^T
#pragma unroll
        for (int kk = 0; kk < 16; ++kk) {
            v2f b = *(const v2f*)(wroot + kk * 4);
            acc = __builtin_amdgcn_wmma_f32_16x16x4_f32(
                false, a_x[kk], false, b, (short)0, acc, false, false);
        }

        const float bias = brel[n0 + l16];
#pragma unroll
        for (int r = 0; r < 8; ++r) {
            float v = acc[r] + bias;
            v = v > 0.f ? v : 0.f;
            out[(size_t)(mbase + r) * D_DIM + n0 + l16] = v;
        }
    }
}

// ---------------------------------------------------------------------------
extern "C" void kernel_launch(void* const* d_in, const int* in_sizes, int n_in,
                              void* d_out, int out_size, void* d_ws, size_t ws_size,
                              hipStream_t stream) {
    const float*        x     = (const float*)d_in[0];
    const unsigned int* eidx  = (const unsigned int*)d_in[1]; // int64 [2,E] viewed as u32 pairs
    const float*        Wrel  = (const float*)d_in[2];
    const float*        brel  = (const float*)d_in[3];
    const float*        Wroot = (const float*)d_in[4];
    float*              out   = (float*)d_out;

    const unsigned int* src_lo = eidx;                 // row 0 (low words at even u32)
    const unsigned int* dst_lo = eidx + 2 * N_EDGES;   // row 1
    float* agg = (float*)d_ws;                         // N_NODES * D_DIM floats = 12.8 MB

    // 1) zero agg
    {
        int n4 = N_NODES * D_DIM / 4;        // 800000 float4
        int blocks = (n4 + 255) / 256;
        gnn_zero_kernel<<<blocks, 256, 0, stream>>>((float4*)agg, n4);
    }
    // 2) scatter-add
    {
        int total = N_EDGES * 16;            // 16 threads per edge
        int blocks = (total + 255) / 256;
        gnn_scatter_kernel<<<blocks, 256, 0, stream>>>(x, src_lo, dst_lo, agg, total);
    }
    // 3) fused dual-GEMM + bias + relu via WMMA (one wave per 16-row strip)
    {
        int blocks = (M_STRIPS + 7) / 8;     // 8 waves per 256-thread block
        gnn_wmma_gemm_kernel<<<blocks, 256, 0, stream>>>(agg, x, Wrel, brel, Wroot, out);
    }
}